// Conv_SNU_Network_classification_66821101191630
// MI455X (gfx1250) — compile-verified
//
#include <hip/hip_runtime.h>
#include <hip/hip_bf16.h>

#define L_TAU   0.8f
#define NUM_T   20
#define BATCH   128
#define IMG     64      // 64x64 input
#define OUTW    55      // conv output 55x55
#define POOLW   27      // pooled 27x27
#define NCH     6
#define FLAT    (6*27*27)   // 4374
#define KSZ     100     // 10x10 kernel

typedef __attribute__((ext_vector_type(2))) float v2f;
typedef __attribute__((ext_vector_type(8))) float v8f;

// ---- workspace layout (floats) ----
#define S1_OFF 0
#define Y1_OFF 2323200
#define S2_OFF 4646400
#define Y2_OFF 4646656
#define WS_FLOATS 4646912

// ---- d_out layout (floats) ----
// loss @0, m @1 (256), out_rec @257 (128*21*2 = 5376), acc @5633
#define OUT_M    1
#define OUT_REC  257
#define OUT_ACC  5633

// x element offset (floats) of kernel tap k: x layout [B,4096,20],
// pixel stride 20 floats, image row stride 64*20 floats.
#define XOFF(k) ((((k) / 10) * IMG + ((k) % 10)) * NUM_T)

__device__ __forceinline__ float sigmoidf_(float z) {
    return 1.0f / (1.0f + __expf(-z));
}

// ------------------------------------------------------------------
// Init: zero SNU state in workspace and out_rec[:, 0, :] in d_out.
// ------------------------------------------------------------------
__global__ __launch_bounds__(256) void init_kernel(float* ws, float* out) {
    size_t idx = (size_t)blockIdx.x * blockDim.x + threadIdx.x;
    if (idx < (size_t)WS_FLOATS) ws[idx] = 0.0f;
    if (idx < 2 * BATCH) {
        int b = (int)(idx >> 1), i = (int)(idx & 1);
        out[OUT_REC + b * (2 * (NUM_T + 1)) + i] = 0.0f;   // t = 0 slot
    }
}

// ------------------------------------------------------------------
// Conv (im2col GEMM via V_WMMA_F32_16X16X4_F32) + SNU layer-1 update.
// grid = B*55 blocks (one output row), 128 threads = 4 waves; wave w
// owns the 16-pixel strip at col 16*w.
//
// K-slot remap: physical slot s of k-group g <-> kernel tap (g + 25*s).
// Slots {2h, 2h+1} of lane-half h then read taps {g+50h, g+25+50h};
// 50 taps == 5 image rows, so 'h' folds into one constant pointer shift
// and every tap load has a compile-time-constant offset.
// Weights are staged into LDS pre-permuted the same way, with a zero
// row so lanes >= NCH stay branch-free.
// ------------------------------------------------------------------
__global__ __launch_bounds__(128) void conv_snu_kernel(
    const float* __restrict__ x, const float* __restrict__ Wc,
    const float* __restrict__ bc, float* __restrict__ ws, int t)
{
    __shared__ float wsh[704];   // 7 rows x 100 (row 6 = zeros) + pad
    for (int i = threadIdx.x; i < 704; i += 128) {
        const int n = i / KSZ;
        const int r = i - n * KSZ;
        const int src = n * KSZ + (r >> 2) + 25 * (r & 3);  // tap = g + 25*s
        wsh[i] = (n < NCH) ? Wc[src] : 0.0f;
    }
    __syncthreads();

    float* s1 = ws + S1_OFF;
    float* y1 = ws + Y1_OFF;

    const int b    = blockIdx.x / OUTW;
    const int row  = blockIdx.x % OUTW;
    const int wave = threadIdx.x >> 5;
    const int lane = threadIdx.x & 31;
    const int half = lane >> 4;
    const int lid  = lane & 15;        // A: pixel M ; B: channel N
    const int col0 = wave * 16;

    // clamp column once -> all tap loads provably in-bounds; clamped lanes
    // only feed masked-out D rows.
    int ci = col0 + lid;
    if (ci > OUTW - 1) ci = OUTW - 1;
    const float* xl = x + ((size_t)b * (IMG * IMG) + (size_t)(row * IMG + ci)) * NUM_T
                        + t + half * (5 * IMG * NUM_T);   // +50 taps for upper half

    const int nrow  = (lid < NCH) ? lid : NCH;   // row 6 is the zero row
    const int wbase = nrow * KSZ + 2 * half;     // even -> 8B-aligned LDS reads

    v8f acc = {};
    #pragma unroll
    for (int g = 0; g < 25; ++g) {
        v2f a;
        a.x = xl[XOFF(g)];          // tap g      + 50*half
        a.y = xl[XOFF(g + 25)];     // tap g + 25 + 50*half
        const v2f w = *(const v2f*)&wsh[wbase + 4 * g];
        acc = __builtin_amdgcn_wmma_f32_16x16x4_f32(
            false, a, false, w, (short)0, acc, false, false);
    }

    // C/D layout: VGPR j -> pixel M = j (lanes 0-15) or j+8 (lanes 16-31),
    // lane -> channel N. SNU: s = relu(c + tau*s*(1-y)); y = sigmoid(s+bc)
    const int ch = lid;
    if (ch < NCH) {
        const float bias = bc[ch];
        const size_t base = ((size_t)b * NCH + ch) * (OUTW * OUTW) + (size_t)row * OUTW;
        #pragma unroll
        for (int j = 0; j < 8; ++j) {
            const int col = col0 + j + 8 * half;
            if (col < OUTW) {
                const size_t idx = base + col;
                const float s  = s1[idx];
                const float yv = y1[idx];
                const float c  = acc[j];
                const float sn = fmaxf(c + L_TAU * s * (1.0f - yv), 0.0f);
                const float yn = sigmoidf_(sn + bias);
                s1[idx] = sn;
                y1[idx] = yn;
            }
        }
    }
}

// ------------------------------------------------------------------
// Maxpool(2x2) + FC (4374 -> 2) + SNU layer-2; writes out_rec[b, t+1, :].
// One block per image, 256 threads, LDS tree reduction.
// ------------------------------------------------------------------
__global__ __launch_bounds__(256) void fc_snu_kernel(
    const float* __restrict__ W2, const float* __restrict__ b2,
    float* __restrict__ ws, float* __restrict__ out, int t)
{
    const float* y1 = ws + Y1_OFF;
    float* s2 = ws + S2_OFF;
    float* y2 = ws + Y2_OFF;

    const int b   = blockIdx.x;
    const int tid = threadIdx.x;

    __shared__ float r0[256];
    __shared__ float r1[256];

    float a0 = 0.0f, a1 = 0.0f;
    for (int i = tid; i < FLAT; i += 256) {
        const int ch  = i / (POOLW * POOLW);
        const int rem = i - ch * (POOLW * POOLW);
        const int pr  = rem / POOLW;
        const int pc  = rem - pr * POOLW;
        const float* yb = y1 + ((size_t)b * NCH + ch) * (OUTW * OUTW)
                             + (size_t)(2 * pr) * OUTW + (2 * pc);
        const float h = fmaxf(fmaxf(yb[0], yb[1]), fmaxf(yb[OUTW], yb[OUTW + 1]));
        a0 += h * W2[(size_t)i * 2 + 0];
        a1 += h * W2[(size_t)i * 2 + 1];
    }
    r0[tid] = a0; r1[tid] = a1;
    __syncthreads();
    for (int s = 128; s > 0; s >>= 1) {
        if (tid < s) { r0[tid] += r0[tid + s]; r1[tid] += r1[tid + s]; }
        __syncthreads();
    }
    if (tid == 0) {
        const float d[2] = { r0[0], r1[0] };
        #pragma unroll
        for (int i = 0; i < 2; ++i) {
            const float s  = s2[b * 2 + i];
            const float yv = y2[b * 2 + i];
            const float sn = fmaxf(d[i] + L_TAU * s * (1.0f - yv), 0.0f);
            const float yn = sigmoidf_(sn + b2[i]);
            s2[b * 2 + i] = sn;
            y2[b * 2 + i] = yn;
            out[OUT_REC + b * (2 * (NUM_T + 1)) + (t + 1) * 2 + i] = yn;
        }
    }
}

// ------------------------------------------------------------------
// Finish: m = sum_t(out_rec)/20, log-softmax CE loss, accuracy.
// Single block, 128 threads (one per image).
// ------------------------------------------------------------------
__global__ __launch_bounds__(128) void finish_kernel(
    const long long* __restrict__ y, float* __restrict__ out)
{
    const int b = threadIdx.x;
    __shared__ float lred[128];
    __shared__ float ared[128];

    const float* rec = out + OUT_REC + b * (2 * (NUM_T + 1));
    float m0 = 0.0f, m1 = 0.0f;
    #pragma unroll
    for (int tt = 0; tt <= NUM_T; ++tt) { m0 += rec[tt * 2]; m1 += rec[tt * 2 + 1]; }
    m0 *= (1.0f / 20.0f);
    m1 *= (1.0f / 20.0f);
    out[OUT_M + b * 2 + 0] = m0;
    out[OUT_M + b * 2 + 1] = m1;

    const float mx  = fmaxf(m0, m1);
    const float lse = mx + __logf(__expf(m0 - mx) + __expf(m1 - mx));
    const int lab = (int)y[b];
    const float logp = ((lab == 1) ? m1 : m0) - lse;
    const int pred = (m1 > m0) ? 1 : 0;
    lred[b] = -logp;
    ared[b] = (pred == lab) ? 1.0f : 0.0f;
    __syncthreads();
    for (int s = 64; s > 0; s >>= 1) {
        if (b < s) { lred[b] += lred[b + s]; ared[b] += ared[b + s]; }
        __syncthreads();
    }
    if (b == 0) {
        out[0]       = lred[0] * (1.0f / BATCH);
        out[OUT_ACC] = ared[0] * (1.0f / BATCH);
    }
}

extern "C" void kernel_launch(void* const* d_in, const int* in_sizes, int n_in,
                              void* d_out, int out_size, void* d_ws, size_t ws_size,
                              hipStream_t stream) {
    const float*     x  = (const float*)d_in[0];
    const long long* y  = (const long long*)d_in[1];   // int64 labels
    const float*     Wc = (const float*)d_in[2];
    const float*     bc = (const float*)d_in[3];
    const float*     W2 = (const float*)d_in[4];
    const float*     b2 = (const float*)d_in[5];
    float* out = (float*)d_out;
    float* ws  = (float*)d_ws;
    (void)in_sizes; (void)n_in; (void)out_size; (void)ws_size;

    const int initBlocks = (WS_FLOATS + 255) / 256;
    init_kernel<<<initBlocks, 256, 0, stream>>>(ws, out);

    for (int t = 0; t < NUM_T; ++t) {
        conv_snu_kernel<<<BATCH * OUTW, 128, 0, stream>>>(x, Wc, bc, ws, t);
        fc_snu_kernel<<<BATCH, 256, 0, stream>>>(W2, b2, ws, out, t);
    }

    finish_kernel<<<1, 128, 0, stream>>>(y, out);
}